// ILRMA_49091476194098
// MI455X (gfx1250) — compile-verified
//
#include <hip/hip_runtime.h>
#include <math.h>

// ---------------------------------------------------------------------------
// ILRMA for MI455X (gfx1250, wave32).
// Sizes fixed by the reference: M=N=2, I=2049, J=2048, K=8, 5 iterations.
// Bandwidth-bound fp32 problem (~2 GB total traffic -> ~90us @ 23.3 TB/s).
// WMMA (V_WMMA_F32_16X16X4_F32) used for the J-reduction GEMMs of the
// T-update: numT = P @ V^T, denT = Rd @ V^T  (16 rows x 8 cols, K-dim = J).
// K1 is parallelized 8x over J segments (129 blocks x 8 waves = 1032 waves)
// with a deterministic LDS combine, so it stays bandwidth-bound instead of
// being occupancy/VALU-bound.
// ---------------------------------------------------------------------------

typedef __attribute__((ext_vector_type(2))) float v2f;
typedef __attribute__((ext_vector_type(8))) float v8f;

#define I_BINS 2049
#define J_FR   2048
#define K_NMF  8
#define EPS_P  1e-20f

// ---------------- complex helpers (match jnp semantics; NO conjugation in
// the covariance / nrm einsums, principal-branch complex sqrt) --------------
struct c32 { float x, y; };
__device__ inline c32 cmk(float a, float b) { c32 r; r.x = a; r.y = b; return r; }
__device__ inline c32 cadd(c32 a, c32 b) { return cmk(a.x + b.x, a.y + b.y); }
__device__ inline c32 csub(c32 a, c32 b) { return cmk(a.x - b.x, a.y - b.y); }
__device__ inline c32 cneg(c32 a)        { return cmk(-a.x, -a.y); }
__device__ inline c32 cmul(c32 a, c32 b) { return cmk(a.x*b.x - a.y*b.y, a.x*b.y + a.y*b.x); }
__device__ inline c32 cscale(c32 a, float s) { return cmk(a.x*s, a.y*s); }
__device__ inline c32 cdiv(c32 a, c32 b) {
  float d = b.x*b.x + b.y*b.y;
  return cmk((a.x*b.x + a.y*b.y)/d, (a.y*b.x - a.x*b.y)/d);
}
__device__ inline c32 csqrt_(c32 z) {          // principal branch
  float r = sqrtf(z.x*z.x + z.y*z.y);
  float u = sqrtf(fmaxf(0.5f*(r + z.x), 0.0f));
  float v = sqrtf(fmaxf(0.5f*(r - z.x), 0.0f));
  if (z.y < 0.0f) v = -v;
  return cmk(u, v);
}

__device__ inline float wave_sum32(float v) {
  #pragma unroll
  for (int off = 16; off > 0; off >>= 1) v += __shfl_xor(v, off, 32);
  return v;
}

// ---------------------------------------------------------------------------
// K0a: transpose X (M,J,I,cplx) -> Xt (I,M,J,cplx); also seed Y = Xc (W = I).
// ---------------------------------------------------------------------------
__global__ __launch_bounds__(256) void k_transpose(const float* __restrict__ X,
                                                   float* __restrict__ Xt,
                                                   float* __restrict__ Y) {
  __shared__ float2 tile[16][17];
  const int m  = blockIdx.z;
  const int i0 = blockIdx.x * 16;
  const int j0 = blockIdx.y * 16;
  const int tx = threadIdx.x, ty = threadIdx.y;

  int i = i0 + tx;          // read: consecutive tx -> consecutive i (coalesced)
  int j = j0 + ty;
  float2 v = make_float2(0.0f, 0.0f);
  if (i < I_BINS) v = *(const float2*)(X + ((size_t)(m * J_FR + j) * I_BINS + i) * 2);
  tile[ty][tx] = v;
  __syncthreads();

  i = i0 + ty;              // write: consecutive tx -> consecutive j (coalesced)
  j = j0 + tx;
  if (i < I_BINS) {
    float2 w = tile[tx][ty];
    size_t o = ((size_t)(i * 2 + m) * J_FR + j) * 2;
    *(float2*)(Xt + o) = w;
    *(float2*)(Y  + o) = w;   // initial Y = Xc since W = identity
  }
}

// ---------------------------------------------------------------------------
// K0b: unpack T0 (I,K,N) -> Tw[n][i][k], V0 (K,J,N) -> Vw[n][k][j], W = I.
// ---------------------------------------------------------------------------
__global__ __launch_bounds__(256) void k_init(const float* __restrict__ T0,
                                              const float* __restrict__ V0,
                                              float* __restrict__ Tw,
                                              float* __restrict__ Tw2,
                                              float* __restrict__ Vw,
                                              float* __restrict__ Wb) {
  const int idx = blockIdx.x * 256 + threadIdx.x;
  if (idx < 2 * I_BINS * K_NMF) {
    int n   = idx / (I_BINS * K_NMF);
    int rem = idx - n * (I_BINS * K_NMF);
    int i = rem >> 3, k = rem & 7;
    float t = T0[(i * K_NMF + k) * 2 + n];
    Tw[idx] = t; Tw2[idx] = t;
  }
  if (idx < 2 * K_NMF * J_FR) {
    int n   = idx / (K_NMF * J_FR);
    int rem = idx - n * (K_NMF * J_FR);
    int k = rem >> 11, j = rem & (J_FR - 1);
    Vw[idx] = V0[(k * J_FR + j) * 2 + n];
  }
  if (idx < I_BINS) {
    float* w = Wb + idx * 8;          // W[i][r][m] complex, r-major
    w[0] = 1.0f; w[1] = 0.0f;         // W[i][0][0] = 1
    w[2] = 0.0f; w[3] = 0.0f;         // W[i][0][1] = 0
    w[4] = 0.0f; w[5] = 0.0f;         // W[i][1][0] = 0
    w[6] = 1.0f; w[7] = 0.0f;         // W[i][1][1] = 1
  }
}

// ---------------------------------------------------------------------------
// K1: T-update via WMMA.  One BLOCK per 16-row tile of I; the block's 8 waves
// each own a 256-column segment of J and accumulate private WMMA C tiles:
//   Rn = T@V (K=8 dot, recomputed on the fly), P = |Y|^2/(Rn^2+eps), Rd = 1/Rn
//   numT(16x8) = P @ V^T, denT(16x8) = Rd @ V^T  via V_WMMA_F32_16X16X4_F32.
// The 8 partial C tiles are combined in LDS in a FIXED order (deterministic,
// graph-replay-safe), then Tnew = Told * sqrt(numT/denT) is written to Tw2
// (Tw keeps old T: K2a still needs it -- reference does not refresh P/Rd).
// WMMA layouts (CDNA5 ISA):
//   A 16x4 f32 : lane L -> row L&15, {VGPR0,VGPR1} = K = 2*(L>>4)+{0,1}
//   B 4x16 f32 : lane L -> col L&15, {VGPR0,VGPR1} = K = 2*(L>>4)+{0,1}
//   C/D 16x16  : lane L -> col L&15, VGPR v -> row v + 8*(L>>4)
// ---------------------------------------------------------------------------
__global__ __launch_bounds__(256) void k1_updateT(const float* __restrict__ Yn,
                                                  const float* __restrict__ TnO,
                                                  const float* __restrict__ Vn,
                                                  float* __restrict__ TnN) {
  __shared__ float red[8][32][16];                // 16 KB partial C tiles

  const int wid  = threadIdx.x >> 5;              // j-segment 0..7 (wave id)
  const int lane = threadIdx.x & 31;
  const int r    = lane & 15;
  const int half = lane >> 4;
  const int i0   = blockIdx.x * 16;               // tile of 16 bins
  const int row  = i0 + r;
  const bool valid = (row < I_BINS);
  const int rowc = valid ? row : (I_BINS - 1);    // clamp: no OOB, no divergence

  float tr[K_NMF];
  #pragma unroll
  for (int k = 0; k < K_NMF; ++k) tr[k] = TnO[rowc * K_NMF + k];

  const float* Yrow = Yn + (size_t)rowc * 4 * J_FR;   // (i*2+n) row of Y, cplx

  v8f cNum = {0.f,0.f,0.f,0.f,0.f,0.f,0.f,0.f};
  v8f cDen = {0.f,0.f,0.f,0.f,0.f,0.f,0.f,0.f};

  const int jb = wid * (J_FR / 8);                // 256-wide segment per wave
  const int je = jb + (J_FR / 8);
  for (int j0 = jb; j0 < je; j0 += 4) {
    const int jA = j0 + (half << 1);              // this lane's K-pair columns

    // |Y|^2 at (rowc, jA) and (rowc, jA+1): one 16B load
    float4 y = *(const float4*)(Yrow + jA * 2);

    // Rn at the two columns (K=8 dot against V)
    float rn0 = 0.0f, rn1 = 0.0f;
    #pragma unroll
    for (int k = 0; k < K_NMF; ++k) {
      float2 vk = *(const float2*)(Vn + k * J_FR + jA);
      rn0 = fmaf(tr[k], vk.x, rn0);
      rn1 = fmaf(tr[k], vk.y, rn1);
    }

    float p0  = (y.x*y.x + y.y*y.y) / (rn0*rn0 + EPS_P);
    float p1  = (y.z*y.z + y.w*y.w) / (rn1*rn1 + EPS_P);
    float rd0 = 1.0f / rn0;
    float rd1 = 1.0f / rn1;
    if (!valid) { p0 = 0.0f; p1 = 0.0f; rd0 = 0.0f; rd1 = 0.0f; }

    v2f Ap; Ap.x = p0;  Ap.y = p1;
    v2f Ar; Ar.x = rd0; Ar.y = rd1;

    // B = V^T chunk: col c = r (k index, zero for c >= 8), rows = same jA pair
    v2f Bv; Bv.x = 0.0f; Bv.y = 0.0f;
    if (r < K_NMF) {
      float2 vb = *(const float2*)(Vn + r * J_FR + jA);
      Bv.x = vb.x; Bv.y = vb.y;
    }

    cNum = __builtin_amdgcn_wmma_f32_16x16x4_f32(false, Ap, false, Bv,
                                                 (short)0, cNum, false, false);
    cDen = __builtin_amdgcn_wmma_f32_16x16x4_f32(false, Ar, false, Bv,
                                                 (short)0, cDen, false, false);
  }

  // deterministic cross-wave combine of the 8 partial C tiles
  #pragma unroll
  for (int v = 0; v < 8; ++v) {
    red[wid][lane][v]     = cNum[v];
    red[wid][lane][v + 8] = cDen[v];
  }
  __syncthreads();

  if (wid == 0) {
    float sNum[8], sDen[8];
    #pragma unroll
    for (int v = 0; v < 8; ++v) { sNum[v] = 0.0f; sDen[v] = 0.0f; }
    #pragma unroll
    for (int w = 0; w < 8; ++w) {                 // fixed summation order
      #pragma unroll
      for (int v = 0; v < 8; ++v) {
        sNum[v] += red[w][lane][v];
        sDen[v] += red[w][lane][v + 8];
      }
    }
    // Scatter: lane holds cols c=r of rows v + 8*half
    if (r < K_NMF) {
      #pragma unroll
      for (int v = 0; v < 8; ++v) {
        const int ri = i0 + v + (half << 3);
        if (ri < I_BINS) {
          const float t = TnO[ri * K_NMF + r];
          TnN[ri * K_NMF + r] = t * sqrtf(sNum[v] / sDen[v]);
        }
      }
    }
  }
}

// ---------------------------------------------------------------------------
// K2a: V-update partials over 16 i-chunks (deterministic two-pass reduction).
// P/Rd from OLD T (reference quirk: not recomputed after T update);
// products use NEW T.  Thread per j column.
// ---------------------------------------------------------------------------
__global__ __launch_bounds__(256) void k2a_partialV(const float* __restrict__ Yn,
                                                    const float* __restrict__ TnO,
                                                    const float* __restrict__ TnN,
                                                    const float* __restrict__ Vn,
                                                    float* __restrict__ part) {
  const int j  = blockIdx.x * 256 + threadIdx.x;
  const int c  = blockIdx.y;
  const int ib = c * 129;
  int ie = ib + 129; if (ie > I_BINS) ie = I_BINS;

  float vj[K_NMF];
  #pragma unroll
  for (int k = 0; k < K_NMF; ++k) vj[k] = Vn[k * J_FR + j];

  float num[K_NMF], den[K_NMF];
  #pragma unroll
  for (int k = 0; k < K_NMF; ++k) { num[k] = 0.0f; den[k] = 0.0f; }

  for (int i = ib; i < ie; ++i) {
    const float* to = TnO + i * K_NMF;            // uniform across block
    const float* tn = TnN + i * K_NMF;
    float2 y = *(const float2*)(Yn + (size_t)i * 4 * J_FR + 2 * j);
    float rn = 0.0f;
    #pragma unroll
    for (int k = 0; k < K_NMF; ++k) rn = fmaf(to[k], vj[k], rn);
    const float P  = (y.x*y.x + y.y*y.y) / (rn*rn + EPS_P);
    const float Rd = 1.0f / rn;
    #pragma unroll
    for (int k = 0; k < K_NMF; ++k) {
      num[k] = fmaf(tn[k], P,  num[k]);
      den[k] = fmaf(tn[k], Rd, den[k]);
    }
  }
  #pragma unroll
  for (int k = 0; k < K_NMF; ++k) {
    part[((size_t)(c * 2 + 0) * K_NMF + k) * J_FR + j] = num[k];
    part[((size_t)(c * 2 + 1) * K_NMF + k) * J_FR + j] = den[k];
  }
}

// K2b: reduce 16 chunks, V *= sqrt(num/den)
__global__ __launch_bounds__(256) void k2b_updateV(const float* __restrict__ part,
                                                   float* __restrict__ Vn) {
  const int j = blockIdx.x * 256 + threadIdx.x;
  #pragma unroll
  for (int k = 0; k < K_NMF; ++k) {
    float num = 0.0f, den = 0.0f;
    #pragma unroll
    for (int c = 0; c < 16; ++c) {
      num += part[((size_t)(c * 2 + 0) * K_NMF + k) * J_FR + j];
      den += part[((size_t)(c * 2 + 1) * K_NMF + k) * J_FR + j];
    }
    Vn[k * J_FR + j] *= sqrtf(num / den);
  }
}

// ---------------------------------------------------------------------------
// K3: per-bin demixing-vector update. One wave per bin i.
//   Rn from NEW T, NEW V.  D = (1/J) sum_j Xc[m]Xc[k]/Rn  (complex SYMMETRIC,
//   no conjugation, matching jnp.einsum).  Solve (W@D) b = e_n (2x2 Cramer),
//   nrm = b.D.b (complex, no conj), W[:, :, n] column <- b / csqrt(nrm).
//   Also commits Tw <- Tw2 for the next round.
// ---------------------------------------------------------------------------
__global__ __launch_bounds__(256) void k3_updateW(const float* __restrict__ Xt,
                                                  const float* __restrict__ TnN,
                                                  const float* __restrict__ Vn,
                                                  float* __restrict__ TnO_out,
                                                  float* __restrict__ Wb,
                                                  int n) {
  const int wid = threadIdx.x >> 5;
  const int i   = blockIdx.x * 8 + wid;
  if (i >= I_BINS) return;
  const int lane = threadIdx.x & 31;

  float tr[K_NMF];
  #pragma unroll
  for (int k = 0; k < K_NMF; ++k) tr[k] = TnN[i * K_NMF + k];

  const float* X0 = Xt + (size_t)i * 4 * J_FR;
  const float* X1 = X0 + 2 * J_FR;

  float d00r = 0.f, d00i = 0.f, d01r = 0.f, d01i = 0.f, d11r = 0.f, d11i = 0.f;
  for (int j = lane; j < J_FR; j += 32) {
    float2 x0 = *(const float2*)(X0 + 2 * j);
    float2 x1 = *(const float2*)(X1 + 2 * j);
    float rn = 0.0f;
    #pragma unroll
    for (int k = 0; k < K_NMF; ++k) rn = fmaf(tr[k], Vn[k * J_FR + j], rn);
    const float s = 1.0f / rn;
    d00r += s * (x0.x*x0.x - x0.y*x0.y);
    d00i += s * (2.0f * x0.x * x0.y);
    d01r += s * (x0.x*x1.x - x0.y*x1.y);
    d01i += s * (x0.x*x1.y + x0.y*x1.x);
    d11r += s * (x1.x*x1.x - x1.y*x1.y);
    d11i += s * (2.0f * x1.x * x1.y);
  }
  d00r = wave_sum32(d00r); d00i = wave_sum32(d00i);
  d01r = wave_sum32(d01r); d01i = wave_sum32(d01i);
  d11r = wave_sum32(d11r); d11i = wave_sum32(d11i);

  if (lane == 0) {
    const float invJ = 1.0f / (float)J_FR;
    c32 d00 = cmk(d00r * invJ, d00i * invJ);
    c32 d01 = cmk(d01r * invJ, d01i * invJ);
    c32 d11 = cmk(d11r * invJ, d11i * invJ);

    float* w = Wb + i * 8;
    c32 w00 = cmk(w[0], w[1]), w01 = cmk(w[2], w[3]);
    c32 w10 = cmk(w[4], w[5]), w11 = cmk(w[6], w[7]);

    // WD = W @ D  (D symmetric: D10 = D01)
    c32 a00 = cadd(cmul(w00, d00), cmul(w01, d01));
    c32 a01 = cadd(cmul(w00, d01), cmul(w01, d11));
    c32 a10 = cadd(cmul(w10, d00), cmul(w11, d01));
    c32 a11 = cadd(cmul(w10, d01), cmul(w11, d11));

    c32 det = csub(cmul(a00, a11), cmul(a01, a10));
    c32 b0, b1;
    if (n == 0) { b0 = cdiv(a11, det);       b1 = cdiv(cneg(a10), det); }
    else        { b0 = cdiv(cneg(a01), det); b1 = cdiv(a00, det); }

    // nrm = b.D.b (no conjugation)
    c32 nrm = cadd(cadd(cmul(cmul(b0, b0), d00),
                        cscale(cmul(cmul(b0, b1), d01), 2.0f)),
                   cmul(cmul(b1, b1), d11));
    c32 sq = csqrt_(nrm);
    c32 o0 = cdiv(b0, sq);
    c32 o1 = cdiv(b1, sq);
    w[0 + n * 2] = o0.x; w[1 + n * 2] = o0.y;   // W[i][0][n]
    w[4 + n * 2] = o1.x; w[5 + n * 2] = o1.y;   // W[i][1][n]
  }
  if (lane < K_NMF) TnO_out[i * K_NMF + lane] = TnN[i * K_NMF + lane];
}

// ---------------------------------------------------------------------------
// K4: full demix Y = W @ Xc, written into d_out (also next iteration's Y).
// ---------------------------------------------------------------------------
__global__ __launch_bounds__(256) void k4_demix(const float* __restrict__ Xt,
                                                const float* __restrict__ Wb,
                                                float* __restrict__ Y) {
  const int i = blockIdx.x;
  const float* w = Wb + i * 8;
  c32 w00 = cmk(w[0], w[1]), w01 = cmk(w[2], w[3]);
  c32 w10 = cmk(w[4], w[5]), w11 = cmk(w[6], w[7]);
  const float* X0 = Xt + (size_t)i * 4 * J_FR;
  const float* X1 = X0 + 2 * J_FR;
  float* Y0 = Y + (size_t)i * 4 * J_FR;
  float* Y1 = Y0 + 2 * J_FR;
  for (int j = threadIdx.x; j < J_FR; j += 256) {
    float2 xv0 = *(const float2*)(X0 + 2 * j);
    float2 xv1 = *(const float2*)(X1 + 2 * j);
    c32 x0 = cmk(xv0.x, xv0.y), x1 = cmk(xv1.x, xv1.y);
    c32 y0 = cadd(cmul(w00, x0), cmul(w01, x1));
    c32 y1 = cadd(cmul(w10, x0), cmul(w11, x1));
    *(float2*)(Y0 + 2 * j) = make_float2(y0.x, y0.y);
    *(float2*)(Y1 + 2 * j) = make_float2(y1.x, y1.y);
  }
}

// ---------------------------------------------------------------------------
extern "C" void kernel_launch(void* const* d_in, const int* in_sizes, int n_in,
                              void* d_out, int out_size, void* d_ws, size_t ws_size,
                              hipStream_t stream) {
  (void)in_sizes; (void)n_in; (void)out_size; (void)ws_size;
  const float* X  = (const float*)d_in[0];   // (2, 2048, 2049, 2) f32
  const float* T0 = (const float*)d_in[1];   // (2049, 8, 2)
  const float* V0 = (const float*)d_in[2];   // (8, 2048, 2)
  float* Y = (float*)d_out;                  // (2049, 2, 2048) complex64 -> f32 pairs

  // workspace layout (floats)
  float* Xt   = (float*)d_ws;                                  // 16,785,408
  float* Tw   = Xt  + (size_t)I_BINS * 2 * J_FR * 2;           // 32,784
  float* Tw2  = Tw  + (size_t)2 * I_BINS * K_NMF;              // 32,784
  float* Vw   = Tw2 + (size_t)2 * I_BINS * K_NMF;              // 32,768
  float* Wb   = Vw  + (size_t)2 * K_NMF * J_FR;                // 16,392
  float* part = Wb  + (size_t)I_BINS * 8;                      // 524,288

  // one-time: transpose X -> Xt, seed Y = Xc, unpack T/V, W = identity
  k_transpose<<<dim3(129, 128, 2), dim3(16, 16), 0, stream>>>(X, Xt, Y);
  k_init<<<129, 256, 0, stream>>>(T0, V0, Tw, Tw2, Vw, Wb);

  const int ntile = (I_BINS + 15) / 16;   // 129

  for (int it = 0; it < 5; ++it) {
    for (int n = 0; n < 2; ++n) {
      const float* Yn = Y + (size_t)n * 2 * J_FR;
      float* TnO = Tw  + (size_t)n * I_BINS * K_NMF;
      float* TnN = Tw2 + (size_t)n * I_BINS * K_NMF;
      float* VnP = Vw  + (size_t)n * K_NMF * J_FR;

      k1_updateT <<<ntile, 256, 0, stream>>>(Yn, TnO, VnP, TnN);
      k2a_partialV<<<dim3(8, 16), 256, 0, stream>>>(Yn, TnO, TnN, VnP, part);
      k2b_updateV<<<8, 256, 0, stream>>>(part, VnP);
      k3_updateW <<<257, 256, 0, stream>>>(Xt, TnN, VnP, TnO, Wb, n);
    }
    k4_demix<<<I_BINS, 256, 0, stream>>>(Xt, Wb, Y);
  }
}